// SelfAttentionLayer_11828339933378
// MI455X (gfx1250) — compile-verified
//
#include <hip/hip_runtime.h>

#define BATCH 4
#define SEQ   4096
#define DDIM  512
#define LDSLD 516            // padded LDS row stride in floats (conflict-free strided reads)
#define NKT   (SEQ / 16)     // 256 key tiles per batch
#define NQT   (SEQ / 16)     // 256 query tiles per batch
#define NROWT ((BATCH * SEQ) / 16)  // 1024 row tiles for projections

typedef float v2f __attribute__((ext_vector_type(2)));
typedef float v8f __attribute__((ext_vector_type(8)));

// D = A(16x4, f32) * B(4x16, f32) + C(16x16, f32) on the CDNA5 matrix pipe
static __device__ __forceinline__ v8f wmma_f32(v2f a, v2f b, v8f c) {
  return __builtin_amdgcn_wmma_f32_16x16x4_f32(
      /*neg_a=*/false, a, /*neg_b=*/false, b,
      /*c_mod=*/(short)0, c, /*reuse_a=*/false, /*reuse_b=*/false);
}

// Asynchronous cooperative copy of a 16 x 512 f32 tile into LDS (padded rows).
// Uses CDNA5 GLOBAL_LOAD_ASYNC_TO_LDS_B128: no VGPR data staging, tracked by
// ASYNCcnt, frees VALU/LDS pipes for compute overlap. GV addressing: 64-bit
// global address in a VGPR pair, wave-relative LDS byte offset in a VGPR
// (low 32 bits of the generic pointer to a __shared__ object).
static __device__ __forceinline__ void copy16xD_async(float* __restrict__ dst,
                                                      const float* __restrict__ src,
                                                      int tid) {
#pragma unroll
  for (int i = 0; i < 8; ++i) {
    int idx = tid + i * 256;       // float4 index 0..2047
    int row = idx >> 7;            // 128 float4 per row
    int c4  = idx & 127;
    unsigned ldsoff = (unsigned)(size_t)(const void*)(dst + row * LDSLD + c4 * 4);
    unsigned long long gaddr =
        (unsigned long long)(size_t)(const void*)(src + row * DDIM + c4 * 4);
    asm volatile("global_load_async_to_lds_b128 %0, %1, off"
                 :: "v"(ldsoff), "v"(gaddr)
                 : "memory");
  }
}

static __device__ __forceinline__ void wait_async() {
  asm volatile("s_wait_asynccnt 0x0" ::: "memory");
}

// Prefetch one 128B line per thread of the next 32KB tile into L2.
static __device__ __forceinline__ void prefetch_tile(const float* p, int tid) {
  __builtin_prefetch(p + tid * 32, 0, 1);
}

// ---------------------------------------------------------------------------
// Kernel 1: fused Q/K/V projection.  y = x @ W^T + b  (torch Linear semantics)
// grid = (1024 row tiles, 3 matrices), block = 256 threads = 8 waves.
// Each wave owns a 64-wide slice of the 512 output columns (4 WMMA N-tiles).
// ---------------------------------------------------------------------------
__global__ __launch_bounds__(256)
void qkv_proj_kernel(const float* __restrict__ x,
                     const float* __restrict__ Wq, const float* __restrict__ bq,
                     const float* __restrict__ Wk, const float* __restrict__ bk,
                     const float* __restrict__ Wv, const float* __restrict__ bv,
                     float* __restrict__ Q, float* __restrict__ K,
                     float* __restrict__ V) {
  __shared__ float xt[16 * LDSLD];               // 32.25 KB staged x tile

  const int tile  = blockIdx.x;                  // row tile over flattened (b,s)
  const int which = blockIdx.y;                  // 0=Q 1=K 2=V
  const float* __restrict__ W    = (which == 0) ? Wq : (which == 1) ? Wk : Wv;
  const float* __restrict__ bias = (which == 0) ? bq : (which == 1) ? bk : bv;
  float* __restrict__ out        = (which == 0) ? Q  : (which == 1) ? K  : V;

  const int row0 = tile * 16;
  const int tid  = threadIdx.x;

  copy16xD_async(xt, x + (size_t)row0 * DDIM, tid);
  wait_async();
  __syncthreads();

  const int wave = tid >> 5;
  const int lane = tid & 31;
  const int half = lane >> 4;                    // 0: lanes 0-15, 1: lanes 16-31
  const int l16  = lane & 15;
  const int ebase = wave * 64;

  v8f acc[4];
#pragma unroll
  for (int t = 0; t < 4; ++t)
#pragma unroll
    for (int r = 0; r < 8; ++r) acc[t][r] = 0.0f;

#pragma unroll 4
  for (int d0 = 0; d0 < DDIM; d0 += 4) {
    const int dA = d0 + half * 2;                // A/B fragment K-layout
    v2f a;
    a.x = xt[l16 * LDSLD + dA];
    a.y = xt[l16 * LDSLD + dA + 1];
#pragma unroll
    for (int t = 0; t < 4; ++t) {
      const int e = ebase + t * 16 + l16;        // B N-index = output col
      v2f b;
      b.x = W[e * DDIM + dA];                    // B[d][n] = W[n][d]
      b.y = W[e * DDIM + dA + 1];
      acc[t] = wmma_f32(a, b, acc[t]);
    }
  }

#pragma unroll
  for (int t = 0; t < 4; ++t) {
    const int n  = ebase + t * 16 + l16;
    const float bn = bias[n];
#pragma unroll
    for (int r = 0; r < 8; ++r) {
      const int m = r + half * 8;                // C layout: rows 0-7 / 8-15
      out[(size_t)(row0 + m) * DDIM + n] = acc[t][r] + bn;
    }
  }
}

// ---------------------------------------------------------------------------
// Kernel 2: flash-style attention (unscaled scores, matching reference).
// grid = (256 query tiles, 4 batches), block = 256 threads = 8 waves.
// Pass 1: online softmax stats (m, l). Pass 2: recompute S, accumulate P*V.
// Waves split D=512 into 64-wide slices; partial S reduced through LDS.
// K/V tiles streamed via GLOBAL_LOAD_ASYNC_TO_LDS_B128 (ASYNCcnt).
// ---------------------------------------------------------------------------
__global__ __launch_bounds__(256)
void attn_kernel(const float* __restrict__ Q, const float* __restrict__ K,
                 const float* __restrict__ V, float* __restrict__ out) {
  __shared__ float kv[16 * LDSLD];               // staged Q/K/V tile (32.25 KB)
  __shared__ float sbuf[8][16][16];              // per-wave partial S (8 KB)
  __shared__ float pbuf[8][16][17];              // P tiles, padded (8.5 KB)

  const int b  = blockIdx.y;
  const int q0 = blockIdx.x * 16;
  const float* __restrict__ Qb = Q + (size_t)b * SEQ * DDIM;
  const float* __restrict__ Kb = K + (size_t)b * SEQ * DDIM;
  const float* __restrict__ Vb = V + (size_t)b * SEQ * DDIM;
  float* __restrict__ Ob       = out + (size_t)b * SEQ * DDIM;

  const int tid  = threadIdx.x;
  const int wave = tid >> 5;
  const int lane = tid & 31;
  const int half = lane >> 4;
  const int l16  = lane & 15;
  const int dsl  = wave * 64;                    // this wave's D slice

  // --- stage Q tile once, keep this wave's A-fragments in registers -------
  copy16xD_async(kv, Qb + (size_t)q0 * DDIM, tid);
  wait_async();
  __syncthreads();
  float qa0[16], qa1[16];
#pragma unroll
  for (int k = 0; k < 16; ++k) {
    const int d = dsl + 4 * k + half * 2;
    qa0[k] = kv[l16 * LDSLD + d];
    qa1[k] = kv[l16 * LDSLD + d + 1];
  }
  __syncthreads();

  float m[8], l[8];
#pragma unroll
  for (int r = 0; r < 8; ++r) { m[r] = -3.0e38f; l[r] = 0.0f; }

  // ===================== Pass 1: softmax statistics =======================
  for (int j = 0; j < NKT; ++j) {
    copy16xD_async(kv, Kb + (size_t)j * 16 * DDIM, tid);
    if (j + 1 < NKT) prefetch_tile(Kb + (size_t)(j + 1) * 16 * DDIM, tid);
    wait_async();
    __syncthreads();

    v8f s;
#pragma unroll
    for (int r = 0; r < 8; ++r) s[r] = 0.0f;
#pragma unroll
    for (int k = 0; k < 16; ++k) {
      const int d = dsl + 4 * k + half * 2;
      v2f a; a.x = qa0[k]; a.y = qa1[k];
      v2f bb;                                    // B[d][n] = K[key n][d]
      bb.x = kv[l16 * LDSLD + d];
      bb.y = kv[l16 * LDSLD + d + 1];
      s = wmma_f32(a, bb, s);
    }
#pragma unroll
    for (int r = 0; r < 8; ++r) sbuf[wave][r + half * 8][l16] = s[r];
    __syncthreads();

#pragma unroll
    for (int r = 0; r < 8; ++r) {
      float v = 0.0f;
#pragma unroll
      for (int w = 0; w < 8; ++w) v += sbuf[w][r + half * 8][l16];
      float t = v;                               // row-max over 16 lanes (half)
      t = fmaxf(t, __shfl_xor(t, 1));
      t = fmaxf(t, __shfl_xor(t, 2));
      t = fmaxf(t, __shfl_xor(t, 4));
      t = fmaxf(t, __shfl_xor(t, 8));
      const float mnew = fmaxf(m[r], t);
      float p = __expf(v - mnew);
      p += __shfl_xor(p, 1);
      p += __shfl_xor(p, 2);
      p += __shfl_xor(p, 4);
      p += __shfl_xor(p, 8);
      l[r] = l[r] * __expf(m[r] - mnew) + p;
      m[r] = mnew;
    }
  }
  __syncthreads();

  // ===================== Pass 2: accumulate P @ V =========================
  v8f o[4];
#pragma unroll
  for (int t = 0; t < 4; ++t)
#pragma unroll
    for (int r = 0; r < 8; ++r) o[t][r] = 0.0f;

  for (int j = 0; j < NKT; ++j) {
    copy16xD_async(kv, Kb + (size_t)j * 16 * DDIM, tid);
    prefetch_tile(Vb + (size_t)j * 16 * DDIM, tid);
    wait_async();
    __syncthreads();

    v8f s;
#pragma unroll
    for (int r = 0; r < 8; ++r) s[r] = 0.0f;
#pragma unroll
    for (int k = 0; k < 16; ++k) {
      const int d = dsl + 4 * k + half * 2;
      v2f a; a.x = qa0[k]; a.y = qa1[k];
      v2f bb;
      bb.x = kv[l16 * LDSLD + d];
      bb.y = kv[l16 * LDSLD + d + 1];
      s = wmma_f32(a, bb, s);
    }
#pragma unroll
    for (int r = 0; r < 8; ++r) sbuf[wave][r + half * 8][l16] = s[r];
    __syncthreads();                             // all K reads done

    // Issue async V-tile copy, then overlap it with the reduction + exp.
    copy16xD_async(kv, Vb + (size_t)j * 16 * DDIM, tid);

#pragma unroll
    for (int r = 0; r < 8; ++r) {
      float v = 0.0f;
#pragma unroll
      for (int w = 0; w < 8; ++w) v += sbuf[w][r + half * 8][l16];
      pbuf[wave][r + half * 8][l16] = __expf(v - m[r]);
    }
    wait_async();
    __syncthreads();                             // V tile + P tile ready

#pragma unroll
    for (int kk = 0; kk < 4; ++kk) {
      const int kc = 4 * kk + half * 2;
      v2f a;                                     // A[m][k] = P[query m][key k]
      a.x = pbuf[wave][l16][kc];
      a.y = pbuf[wave][l16][kc + 1];
#pragma unroll
      for (int t = 0; t < 4; ++t) {
        const int n = dsl + t * 16 + l16;
        v2f bb;                                  // B[k][n] = V[key k][col n]
        bb.x = kv[kc * LDSLD + n];
        bb.y = kv[(kc + 1) * LDSLD + n];
        o[t] = wmma_f32(a, bb, o[t]);
      }
    }
    __syncthreads();                             // protect kv before next K copy
  }

  float invl[8];
#pragma unroll
  for (int r = 0; r < 8; ++r) invl[r] = 1.0f / l[r];

#pragma unroll
  for (int t = 0; t < 4; ++t) {
    const int n = dsl + t * 16 + l16;
#pragma unroll
    for (int r = 0; r < 8; ++r) {
      const int mm = r + half * 8;
      Ob[(size_t)(q0 + mm) * DDIM + n] = o[t][r] * invl[r];
    }
  }
}

extern "C" void kernel_launch(void* const* d_in, const int* in_sizes, int n_in,
                              void* d_out, int out_size, void* d_ws, size_t ws_size,
                              hipStream_t stream) {
  const float* x  = (const float*)d_in[0];
  const float* Wq = (const float*)d_in[1];
  const float* bq = (const float*)d_in[2];
  const float* Wk = (const float*)d_in[3];
  const float* bk = (const float*)d_in[4];
  const float* Wv = (const float*)d_in[5];
  const float* bv = (const float*)d_in[6];
  float* out = (float*)d_out;

  const size_t n = (size_t)BATCH * SEQ * DDIM;   // 8,388,608 elems per tensor
  float* Qw = (float*)d_ws;
  float* Kw = Qw + n;
  float* Vw = Kw + n;

  dim3 g1(NROWT, 3);
  qkv_proj_kernel<<<g1, 256, 0, stream>>>(x, Wq, bq, Wk, bk, Wv, bv, Qw, Kw, Vw);

  dim3 g2(NQT, BATCH);
  attn_kernel<<<g2, 256, 0, stream>>>(Qw, Kw, Vw, out);
}